// LorentzTransformerEncoder_57836029608088
// MI455X (gfx1250) — compile-verified
//
#include <hip/hip_runtime.h>
#include <hip/hip_bf16.h>

// ---------------- problem constants ----------------
#define BATCH   16
#define NTOK    577
#define LFULL   513
#define DMODEL  512
#define HEADS   8
#define HDIM    64
#define MROWS   (BATCH*NTOK)     // 9232 == 577*16
#define KPAD    516              // 513 padded to multiple of 4
#define WMAT    (KPAD*DMODEL)    // padded weight matrix elements
#define KTILES  37               // ceil(577/16)
#define LOG2F_      0.69314718055994531f
#define SP_NEG_M_   0.64439666f  // log1p(exp(-0.1))

typedef __attribute__((ext_vector_type(2))) float v2f;
typedef __attribute__((ext_vector_type(8))) float v8f;
typedef __attribute__((ext_vector_type(4))) int   v4i;

#define AS1 __attribute__((address_space(1)))
#define AS3 __attribute__((address_space(3)))

#if defined(__has_builtin)
#  if __has_builtin(__builtin_amdgcn_global_load_async_to_lds_b128)
#    define HAVE_ASYNC_LDS 1
#  else
#    define HAVE_ASYNC_LDS 0
#  endif
#  if __has_builtin(__builtin_amdgcn_s_wait_asynccnt)
#    define HAVE_WAIT_ASYNC 1
#  else
#    define HAVE_WAIT_ASYNC 0
#  endif
#else
#  define HAVE_ASYNC_LDS 0
#  define HAVE_WAIT_ASYNC 0
#endif

#if HAVE_ASYNC_LDS
static __device__ __forceinline__ void async_cp16(const float* g, float* l) {
    // builtin signature (from clang diagnostic): (AS1 v4i*, AS3 v4i*, imm offset, imm cpol)
    __builtin_amdgcn_global_load_async_to_lds_b128(
        (AS1 v4i*)(g), (AS3 v4i*)(l), /*offset=*/0, /*cpol=*/0);
}
#endif

template <int N>
static __device__ __forceinline__ void wait_asynccnt() {
#if HAVE_WAIT_ASYNC
    __builtin_amdgcn_s_wait_asynccnt(N);
#elif HAVE_ASYNC_LDS
    asm volatile("s_wait_asynccnt %0" :: "i"(N) : "memory");
#endif
}

// D = A(16x4 f32) * B(4x16 f32) + C(16x16 f32)   -- native CDNA5 fp32 WMMA
static __device__ __forceinline__ v8f wmma4(v2f a, v2f b, v8f c) {
    return __builtin_amdgcn_wmma_f32_16x16x4_f32(
        false, a, false, b, (short)0, c, false, false);
}
static __device__ __forceinline__ v8f v8zero() {
    v8f z;
#pragma unroll
    for (int i = 0; i < 8; ++i) z[i] = 0.f;
    return z;
}
static __device__ __forceinline__ float hred_add(float v) {
    v += __shfl_xor(v, 1, 32); v += __shfl_xor(v, 2, 32);
    v += __shfl_xor(v, 4, 32); v += __shfl_xor(v, 8, 32);
    return v;
}
static __device__ __forceinline__ float hred_max(float v) {
    v = fmaxf(v, __shfl_xor(v, 1, 32)); v = fmaxf(v, __shfl_xor(v, 2, 32));
    v = fmaxf(v, __shfl_xor(v, 4, 32)); v = fmaxf(v, __shfl_xor(v, 8, 32));
    return v;
}
static __device__ __forceinline__ float softplusf(float x) {
    return (x > 20.f) ? x : __logf(1.f + __expf(x));
}
static __device__ __forceinline__ float acoshf_(float x) {   // x >= 1
    return __logf(x + sqrtf(fmaxf(x * x - 1.f, 0.f)));
}

// =====================================================================
// Stage 0: pad W{q,k,v,o} (513x512) to 516x512 with zero rows so the
// GEMM K-loops are branch-free (no EXEC-mask tail).
// =====================================================================
__global__ __launch_bounds__(256) void lorentz_pad_w(
    const float* __restrict__ Wq, const float* __restrict__ Wk,
    const float* __restrict__ Wv, const float* __restrict__ Wo,
    float* __restrict__ Wp)
{
    int i = blockIdx.x * 256 + threadIdx.x;
    if (i >= WMAT) return;
    const bool inr = i < LFULL * DMODEL;
    Wp[0 * WMAT + i] = inr ? Wq[i] : 0.f;
    Wp[1 * WMAT + i] = inr ? Wk[i] : 0.f;
    Wp[2 * WMAT + i] = inr ? Wv[i] : 0.f;
    Wp[3 * WMAT + i] = inr ? Wo[i] : 0.f;
}

// =====================================================================
// Stage 1: fused QKV projection, 2x M register blocking.
// block = 256 thr = 8 waves; wave w owns head w (64 cols) for TWO
// 16-row tiles (shared B fragments -> half the weight traffic).
// =====================================================================
__global__ __launch_bounds__(256) void lorentz_qkv_proj(
    const float* __restrict__ x, const float* __restrict__ Wp,  // padded 516x512
    float* __restrict__ outS, float* __restrict__ outT, int is_v)
{
    __shared__ float a_lds[32 * 520];

    const int tid  = threadIdx.x;
    const int lane = tid & 31;
    const int wid  = tid >> 5;                 // head index
    const int half = lane >> 4;
    const int l16  = lane & 15;
    const int m0   = blockIdx.x * 32;

    for (int idx = tid; idx < 32 * 520; idx += 256) {
        int r = idx / 520, c = idx - r * 520;
        int row = m0 + r; if (row > MROWS - 1) row = MROWS - 1;
        a_lds[idx] = (c < LFULL) ? x[(size_t)row * LFULL + c] : 0.f;
    }
    __syncthreads();

    v8f acc[2][4];
#pragma unroll
    for (int tt = 0; tt < 2; ++tt)
#pragma unroll
        for (int nt = 0; nt < 4; ++nt) acc[tt][nt] = v8zero();

    for (int kc = 0; kc < 129; ++kc) {
        const int k = kc * 4 + 2 * half;
        v2f a0, a1;
        a0.x = a_lds[l16 * 520 + k];        a0.y = a_lds[l16 * 520 + k + 1];
        a1.x = a_lds[(16 + l16) * 520 + k]; a1.y = a_lds[(16 + l16) * 520 + k + 1];
#pragma unroll
        for (int nt = 0; nt < 4; ++nt) {
            const int col = wid * HDIM + nt * 16 + l16;
            v2f b;
            b.x = Wp[(size_t)k * DMODEL + col];
            b.y = Wp[(size_t)(k + 1) * DMODEL + col];
            acc[0][nt] = wmma4(a0, b, acc[0][nt]);
            acc[1][nt] = wmma4(a1, b, acc[1][nt]);
        }
    }

#pragma unroll
    for (int tt = 0; tt < 2; ++tt) {
        float rowss[8];
#pragma unroll
        for (int i = 0; i < 8; ++i) {
            float s = 0.f;
#pragma unroll
            for (int nt = 0; nt < 4; ++nt) { float v = acc[tt][nt][i]; s += v * v; }
            rowss[i] = hred_add(s);
        }
        if (!is_v) {
            if (l16 == 0) {
#pragma unroll
                for (int i = 0; i < 8; ++i) {
                    int row = m0 + tt * 16 + half * 8 + i;
                    if (row < MROWS) {
                        int b = row / NTOK, n = row - b * NTOK;
                        outT[((size_t)(b * HEADS + wid)) * NTOK + n] = sqrtf(1.f + rowss[i]);
                    }
                }
            }
#pragma unroll
            for (int i = 0; i < 8; ++i) {
                int row = m0 + tt * 16 + half * 8 + i;
                if (row < MROWS) {
                    int b = row / NTOK, n = row - b * NTOK;
                    float* dst = outS + (((size_t)(b * HEADS + wid)) * NTOK + n) * HDIM;
#pragma unroll
                    for (int nt = 0; nt < 4; ++nt) dst[nt * 16 + l16] = acc[tt][nt][i];
                }
            }
        } else {
            // logmap0: f = arccosh(max(t,1+1e-7)) / max(||s||,1e-8); store f*s
#pragma unroll
            for (int i = 0; i < 8; ++i) {
                float ss = rowss[i];
                float t  = sqrtf(1.f + ss);
                float sn = fmaxf(sqrtf(ss), 1e-8f);
                float f  = acoshf_(fmaxf(t, 1.f + 1e-7f)) / sn;
                int row = m0 + tt * 16 + half * 8 + i;
                if (row < MROWS) {
                    int b = row / NTOK, n = row - b * NTOK;
                    float* dst = outS + (((size_t)(b * HEADS + wid)) * NTOK + n) * HDIM;
#pragma unroll
                    for (int nt = 0; nt < 4; ++nt) dst[nt * 16 + l16] = f * acc[tt][nt][i];
                }
            }
        }
    }
}

// =====================================================================
// Stage 2: fused hyperbolic flash-attention with double-buffered
// async-to-LDS key/value tile streaming (ASYNCcnt path on gfx1250).
// grid(B*H, 5), block 256 = 8 waves; wave handles one 16-query tile.
// =====================================================================
__global__ __launch_bounds__(256) void lorentz_flash_attn(
    const float* __restrict__ qs, const float* __restrict__ ks,
    const float* __restrict__ vts,
    const float* __restrict__ qtime, const float* __restrict__ ktime,
    const float* __restrict__ beta_raw, const float* __restrict__ tau_raw,
    const float* __restrict__ lambda_raw, const float* __restrict__ temperature,
    float* __restrict__ attn_out)
{
    __shared__ float s_k[2][16 * HDIM];
    __shared__ float s_v[2][16 * HDIM];
    __shared__ float s_p[8][16 * 16];

    const int tid  = threadIdx.x;
    const int lane = tid & 31;
    const int wid  = tid >> 5;
    const int half = lane >> 4;
    const int l16  = lane & 15;

    const int bh = blockIdx.x;
    const int b  = bh >> 3;
    const int h  = bh & 7;
    int qt0 = blockIdx.y * 8 + wid;
    if (qt0 > KTILES - 1) qt0 = KTILES - 1;   // clamped dup work, benign

    const float beta = softplusf(beta_raw[0]);
    const float tau  = softplusf(tau_raw[0]);
    const float lam  = softplusf(lambda_raw[0]);
    const float invT = 1.f / temperature[0];

    const float* qs_bh = qs   + ((size_t)bh * NTOK) * HDIM;
    const float* ks_bh = ks   + ((size_t)bh * NTOK) * HDIM;
    const float* vs_bh = vts  + ((size_t)bh * NTOK) * HDIM;
    const float* kt_bh = ktime + (size_t)bh * NTOK;

    // staging coords: one 16B transfer per thread per buffer
    const int srow  = tid >> 4;   // 0..15 key row in tile
    const int squad = tid & 15;   // 0..15 float4 within row

    // preload A fragments (16x64 q_s tile) for all 16 K-chunks
    v2f afrag[16];
    {
        int qr = qt0 * 16 + l16; if (qr > NTOK - 1) qr = NTOK - 1;
#pragma unroll
        for (int kc = 0; kc < 16; ++kc) {
            int d = kc * 4 + 2 * half;
            afrag[kc].x = qs_bh[(size_t)qr * HDIM + d];
            afrag[kc].y = qs_bh[(size_t)qr * HDIM + d + 1];
        }
    }
    // per-row (this half-wave's 8 rows) q_t and B_gate
    float qtv[8], Bg[8];
#pragma unroll
    for (int i = 0; i < 8; ++i) {
        int qr = qt0 * 16 + half * 8 + i; if (qr > NTOK - 1) qr = NTOK - 1;
        float qt_ = qtime[(size_t)bh * NTOK + qr];
        qtv[i] = qt_;
        float ct   = acoshf_(fmaxf(qt_, 1.001f));
        float sb   = beta / sinhf(ct);
        float argB = 1.f - sb * sb;
        Bg[i] = sqrtf(softplusf(4.f * argB) * 0.25f + 1e-8f);
    }

    float mrow[8], lrow[8];
    v8f accv[4];
#pragma unroll
    for (int i = 0; i < 8; ++i) { mrow[i] = -3.0e38f; lrow[i] = 0.f; }
#pragma unroll
    for (int nt = 0; nt < 4; ++nt) accv[nt] = v8zero();

#if HAVE_ASYNC_LDS
    {   // prologue: async-stage tile 0 into buffer 0
        int kr = 0 * 16 + srow; if (kr > NTOK - 1) kr = NTOK - 1;
        async_cp16(ks_bh + (size_t)kr * HDIM + squad * 4, &s_k[0][srow * HDIM + squad * 4]);
        async_cp16(vs_bh + (size_t)kr * HDIM + squad * 4, &s_v[0][srow * HDIM + squad * 4]);
    }
#endif

    for (int t = 0; t < KTILES; ++t) {
        const int cur = t & 1;
#if HAVE_ASYNC_LDS
        if (t + 1 < KTILES) {
            int kr = (t + 1) * 16 + srow; if (kr > NTOK - 1) kr = NTOK - 1;
            async_cp16(ks_bh + (size_t)kr * HDIM + squad * 4, &s_k[1 - cur][srow * HDIM + squad * 4]);
            async_cp16(vs_bh + (size_t)kr * HDIM + squad * 4, &s_v[1 - cur][srow * HDIM + squad * 4]);
            wait_asynccnt<2>();   // current tile's transfers done; next still in flight
        } else {
            wait_asynccnt<0>();
        }
        __syncthreads();
#else
        {
            int kr = t * 16 + srow; if (kr > NTOK - 1) kr = NTOK - 1;
            const float4 kk = *(const float4*)(ks_bh + (size_t)kr * HDIM + squad * 4);
            const float4 vv = *(const float4*)(vs_bh + (size_t)kr * HDIM + squad * 4);
            *(float4*)&s_k[cur][srow * HDIM + squad * 4] = kk;
            *(float4*)&s_v[cur][srow * HDIM + squad * 4] = vv;
        }
        __syncthreads();
#endif

        // ---- QK^T : 16 WMMAs over K=64 ----
        v8f cqk = v8zero();
#pragma unroll
        for (int kc = 0; kc < 16; ++kc) {
            int d = kc * 4 + 2 * half;
            v2f bf; bf.x = s_k[cur][l16 * HDIM + d]; bf.y = s_k[cur][l16 * HDIM + d + 1];
            cqk = wmma4(afrag[kc], bf, cqk);
        }

        // ---- hyperbolic epilogue ----
        int keyi = t * 16 + l16;
        const bool keyvalid = keyi < NTOK;
        if (keyi > NTOK - 1) keyi = NTOK - 1;
        const float kti = kt_bh[keyi];
        float logit[8];
#pragma unroll
        for (int i = 0; i < 8; ++i) {
            float inner  = cqk[i] - qtv[i] * kti;           // -q_t k_t + q_s.k_s
            float nsQK   = fmaxf(inner * inner - 1.f, 0.f);
            float nsOQ   = fmaxf(qtv[i] * qtv[i] - 1.f, 0.f);
            float numerZ = -kti - inner * qtv[i];
            float Zraw   = numerZ * rsqrtf(nsQK + 1e-12f) * rsqrtf(nsOQ + 1e-12f);
            bool  valid  = (nsQK > 1e-6f) && (nsOQ > 1e-6f) && !(Zraw != Zraw);
            float Zsafe  = valid ? fminf(fmaxf(Zraw, -1.f), 1.f) : 0.f;
            float dist   = acoshf_(fmaxf(-inner, 1.001f));
            float sd     = 40.f * tanhf(dist * (1.f / 40.f)) * (1.f / 15.f);
            float Hv     = sd + softplusf(-2.f * sd) - LOG2F_;
            float Phi    = softplusf(Bg[i] + Zsafe - 0.1f) - SP_NEG_M_;
            float lg     = (-lam * Hv - tau * Phi) * invT;
            logit[i]     = keyvalid ? lg : -3.0e38f;
        }

        // ---- online softmax + p-tile into LDS ----
        float scale[8];
#pragma unroll
        for (int i = 0; i < 8; ++i) {
            float tmax = hred_max(logit[i]);
            float mn   = fmaxf(mrow[i], tmax);
            float pi   = __expf(logit[i] - mn);
            float sc   = __expf(mrow[i] - mn);
            float psum = hred_add(pi);
            lrow[i]  = lrow[i] * sc + psum;
            mrow[i]  = mn;
            scale[i] = sc;
            s_p[wid][(half * 8 + i) * 16 + l16] = pi;
        }
#pragma unroll
        for (int nt = 0; nt < 4; ++nt)
#pragma unroll
            for (int i = 0; i < 8; ++i) accv[nt][i] *= scale[i];

        // ---- p @ v : 16 WMMAs (K=16 keys, N=64 dims) ----
#pragma unroll
        for (int kc = 0; kc < 4; ++kc) {
            int kk = kc * 4 + 2 * half;
            v2f ap; ap.x = s_p[wid][l16 * 16 + kk]; ap.y = s_p[wid][l16 * 16 + kk + 1];
#pragma unroll
            for (int nt = 0; nt < 4; ++nt) {
                v2f bv;
                bv.x = s_v[cur][kk * HDIM + nt * 16 + l16];
                bv.y = s_v[cur][(kk + 1) * HDIM + nt * 16 + l16];
                accv[nt] = wmma4(ap, bv, accv[nt]);
            }
        }
#if HAVE_ASYNC_LDS
        __syncthreads();   // all waves done with buf `cur` before it is re-filled
#endif
    }

    // ---- normalize, expmap0, transposed writeout ----
    const float inv1p = 1.f / (1.f + 1e-8f);
    float meanv[4][8], rss[8];
#pragma unroll
    for (int i = 0; i < 8; ++i) {
        float linv = (lrow[i] > 0.f) ? (1.f / lrow[i]) : 0.f;
        float s = 0.f;
#pragma unroll
        for (int nt = 0; nt < 4; ++nt) {
            float mv = accv[nt][i] * linv * inv1p;
            meanv[nt][i] = mv; s += mv * mv;
        }
        rss[i] = hred_add(s);
    }
#pragma unroll
    for (int i = 0; i < 8; ++i) {
        float sn = fmaxf(sqrtf(rss[i]), 1e-8f);
        float sc = sinhf(sn) / sn;
        int n = qt0 * 16 + half * 8 + i;
        if (n < NTOK) {
            size_t base = ((size_t)(b * NTOK + n)) * DMODEL + h * HDIM;
#pragma unroll
            for (int nt = 0; nt < 4; ++nt)
                attn_out[base + nt * 16 + l16] = sc * meanv[nt][i];
        }
    }
}

// =====================================================================
// Stage 3: output projection (padded Wo -> branch-free K loop).
// full = [time_fix(space), space] (513), out_s = full @ Wo,
// out = [time_fix(out_s), out_s].
// =====================================================================
__global__ __launch_bounds__(256) void lorentz_out_proj(
    const float* __restrict__ attn_space, const float* __restrict__ Wpo,
    float* __restrict__ out)
{
    __shared__ float a_lds[16 * 520];
    __shared__ float rowacc[16];

    const int tid  = threadIdx.x;
    const int lane = tid & 31;
    const int wid  = tid >> 5;
    const int half = lane >> 4;
    const int l16  = lane & 15;
    const int m0   = blockIdx.x * 16;

    if (tid < 16) rowacc[tid] = 0.f;
    __syncthreads();

    {   // load 16x512 space tile; accumulate row norms via ds_add_f32
        int r = tid >> 4, c0 = tid & 15;
        float ssum = 0.f;
        for (int j = 0; j < 32; ++j) {
            int c = c0 + j * 16;
            float v = attn_space[(size_t)(m0 + r) * DMODEL + c];
            a_lds[r * 520 + 1 + c] = v;
            ssum += v * v;
        }
        atomicAdd(&rowacc[r], ssum);
        if (c0 < 7) a_lds[r * 520 + 513 + c0] = 0.f;   // zero K padding 513..519
    }
    __syncthreads();
    if (tid < 16) {
        a_lds[tid * 520 + 0] = sqrtf(1.f + rowacc[tid]);  // full[0] = time fix
        rowacc[tid] = 0.f;                                 // reuse for out norms
    }
    __syncthreads();

    v8f acc[4];
#pragma unroll
    for (int nt = 0; nt < 4; ++nt) acc[nt] = v8zero();

    for (int kc = 0; kc < 129; ++kc) {
        const int k = kc * 4 + 2 * half;
        v2f a; a.x = a_lds[l16 * 520 + k]; a.y = a_lds[l16 * 520 + k + 1];
#pragma unroll
        for (int nt = 0; nt < 4; ++nt) {
            const int col = wid * HDIM + nt * 16 + l16;
            v2f bw;
            bw.x = Wpo[(size_t)k * DMODEL + col];
            bw.y = Wpo[(size_t)(k + 1) * DMODEL + col];
            acc[nt] = wmma4(a, bw, acc[nt]);
        }
    }

    float rowss[8];
#pragma unroll
    for (int i = 0; i < 8; ++i) {
        float s = 0.f;
#pragma unroll
        for (int nt = 0; nt < 4; ++nt) { float v = acc[nt][i]; s += v * v; }
        rowss[i] = hred_add(s);
    }
    if (l16 == 0) {
#pragma unroll
        for (int i = 0; i < 8; ++i) atomicAdd(&rowacc[half * 8 + i], rowss[i]);
    }
#pragma unroll
    for (int i = 0; i < 8; ++i) {
        size_t row = (size_t)(m0 + half * 8 + i);
#pragma unroll
        for (int nt = 0; nt < 4; ++nt)
            out[row * LFULL + 1 + wid * HDIM + nt * 16 + l16] = acc[nt][i];
    }
    __syncthreads();
    if (tid < 16)
        out[(size_t)(m0 + tid) * LFULL] = sqrtf(1.f + rowacc[tid]);
}

// =====================================================================
extern "C" void kernel_launch(void* const* d_in, const int* in_sizes, int n_in,
                              void* d_out, int out_size, void* d_ws, size_t ws_size,
                              hipStream_t stream)
{
    const float* x    = (const float*)d_in[0];
    const float* Wq   = (const float*)d_in[1];
    const float* Wk   = (const float*)d_in[2];
    const float* Wv   = (const float*)d_in[3];
    const float* Wo   = (const float*)d_in[4];
    const float* temp = (const float*)d_in[5];
    const float* beta = (const float*)d_in[6];
    const float* tau  = (const float*)d_in[7];
    const float* lamb = (const float*)d_in[8];
    float* out = (float*)d_out;

    const size_t nBHN64 = (size_t)BATCH * HEADS * NTOK * HDIM;   // 4,726,784
    const size_t nBHN   = (size_t)BATCH * HEADS * NTOK;          // 73,856
    float* qs   = (float*)d_ws;
    float* ks   = qs  + nBHN64;
    float* vts  = ks  + nBHN64;
    float* attn = vts + nBHN64;
    float* qt   = attn + (size_t)MROWS * DMODEL;
    float* kt   = qt + nBHN;
    float* Wp   = kt + nBHN;          // 4 padded matrices, 4*516*512 floats

    lorentz_pad_w<<<dim3((WMAT + 255) / 256), 256, 0, stream>>>(Wq, Wk, Wv, Wo, Wp);

    const int g1 = (MROWS + 31) / 32;   // 289
    lorentz_qkv_proj<<<dim3(g1), 256, 0, stream>>>(x, Wp + 0 * WMAT, qs,  qt, 0);
    lorentz_qkv_proj<<<dim3(g1), 256, 0, stream>>>(x, Wp + 1 * WMAT, ks,  kt, 0);
    lorentz_qkv_proj<<<dim3(g1), 256, 0, stream>>>(x, Wp + 2 * WMAT, vts, qt, 1); // outT unused

    lorentz_flash_attn<<<dim3(BATCH * HEADS, 5), 256, 0, stream>>>(
        qs, ks, vts, qt, kt, beta, tau, lamb, temp, attn);

    lorentz_out_proj<<<dim3(MROWS / 16), 256, 0, stream>>>(attn, Wp + 3 * WMAT, out);
}